// RuleSelector_60327110640326
// MI455X (gfx1250) — compile-verified
//
#include <hip/hip_runtime.h>

typedef __attribute__((ext_vector_type(2))) float v2f;
typedef __attribute__((ext_vector_type(8))) float v8f;

#define W_MODE      1.0f
#define W_COMFORT   0.1f
#define W_PROGRESS  0.5f
#define W_COLLISION 1.0f
#define W_DRIVABLE  0.3f
#define COLL_R2     4.0f     // 2.0^2
#define LANE_R2     9.0f     // 3.0^2
#define MASK_BIG    1.0e12f

#define B_   16
#define M_   60
#define T_   80
#define NAG  128
#define NLN  64
#define NPT  20
#define NLP  (NLN * NPT)     // 1280

#define EGO_TILES   (T_ / 16)    // 5
#define AG_TILES    (NAG / 16)   // 8
#define LN_TILES    (NLP / 16)   // 80

// Single-instruction float min (skip clang's sNaN canonicalization: WMMA
// squared distances are never sNaN here).
__device__ __forceinline__ float fmin_fast(float a, float b) {
    float d;
    asm("v_min_num_f32 %0, %1, %2" : "=v"(d) : "v"(a), "v"(b));
    return d;
}

// One wave (32 lanes) scores one (batch, mode) pair.
// Squared pairwise distances on the matrix pipe via rank-4 factorization:
//   A row  (ego point e):  [ ex, ey ]                       (lanes 0-15, K=0..1)
//                          [ ex^2+ey^2, 1 ]                 (lanes 16-31, K=2..3)
//   B col  (target p):     [ -2px, -2py ]                   (lanes 0-15)
//                          [ 1, px^2+py^2+bigIfMasked ]     (lanes 16-31)
//   C = A x B  ->  C[m][n] = ||e_m - p_n||^2 (+BIG if masked)
__global__ __launch_bounds__(128)
void score_modes_kernel(const float* __restrict__ trajs,       // B,M,T,3
                        const float* __restrict__ agents,      // B,NAG,4
                        const float* __restrict__ agents_mask, // B,NAG
                        const float* __restrict__ lanes,       // B,NLN,NPT,3
                        const float* __restrict__ lanes_mask,  // B,NLN
                        float* __restrict__ partial)           // B*M
{
    const int lane = threadIdx.x & 31;
    const int wave = threadIdx.x >> 5;
    const int idx  = blockIdx.x * 4 + wave;      // 0..959
    const int b    = idx / M_;
    const int m    = idx % M_;
    const float* tr = trajs + (size_t)(b * M_ + m) * (T_ * 3);

    const int  r  = lane & 15;
    const bool hi = (lane >= 16);

    // ---- build the 5 ego A-tiles once (registers) ----
    v2f a[EGO_TILES];
    #pragma unroll
    for (int t = 0; t < EGO_TILES; ++t) {
        const int p = t * 16 + r;
        const float ex = tr[p * 3 + 0];
        const float ey = tr[p * 3 + 1];
        a[t].x = hi ? (ex * ex + ey * ey) : ex;
        a[t].y = hi ? 1.0f : ey;
    }

    v8f mn[EGO_TILES];

    // ================= collision term (agents) =================
    #pragma unroll
    for (int t = 0; t < EGO_TILES; ++t)
        #pragma unroll
        for (int v = 0; v < 8; ++v) mn[t][v] = 3.0e38f;

    for (int j = 0; j < AG_TILES; ++j) {
        const int n = j * 16 + r;
        const v2f axy   = *(const v2f*)(agents + (size_t)(b * NAG + n) * 4);
        const float msk = agents_mask[b * NAG + n];
        v2f bb;
        bb.x = hi ? 1.0f : (-2.0f * axy.x);
        bb.y = hi ? (axy.x * axy.x + axy.y * axy.y + (1.0f - msk) * MASK_BIG)
                  : (-2.0f * axy.y);
        #pragma unroll
        for (int t = 0; t < EGO_TILES; ++t) {
            v8f c = {};
            c = __builtin_amdgcn_wmma_f32_16x16x4_f32(
                    false, a[t], false, bb, (short)0, c, false, false);
            #pragma unroll
            for (int v = 0; v < 8; ++v) mn[t][v] = fmin_fast(mn[t][v], c[v]);
        }
    }

    // Row has a collision iff SOME agent is within radius: OR-reduce a packed
    // 8-bit predicate (one int) across the 16-lane halves, then popcount.
    int coll_rows = 0;
    #pragma unroll
    for (int t = 0; t < EGO_TILES; ++t) {
        unsigned u = 0u;
        #pragma unroll
        for (int v = 0; v < 8; ++v)
            u |= (mn[t][v] < COLL_R2) ? (1u << v) : 0u;
        #pragma unroll
        for (int s = 1; s <= 8; s <<= 1)
            u |= (unsigned)__shfl_xor((int)u, s, 32);
        int cnt = __popc(u);                 // lanes 0-15: rows 0-7; hi: 8-15
        cnt += __shfl_xor(cnt, 16, 32);      // combine the two row-halves
        coll_rows += cnt;
    }

    // ================= drivable term (lane points) =================
    #pragma unroll
    for (int t = 0; t < EGO_TILES; ++t)
        #pragma unroll
        for (int v = 0; v < 8; ++v) mn[t][v] = 3.0e38f;

    for (int j = 0; j < LN_TILES; ++j) {
        const int q  = j * 16 + r;                 // flat lane-point index
        const float lx  = lanes[(size_t)(b * NLP + q) * 3 + 0];
        const float ly  = lanes[(size_t)(b * NLP + q) * 3 + 1];
        const float msk = lanes_mask[b * NLN + q / NPT];
        v2f bb;
        bb.x = hi ? 1.0f : (-2.0f * lx);
        bb.y = hi ? (lx * lx + ly * ly + (1.0f - msk) * MASK_BIG)
                  : (-2.0f * ly);
        #pragma unroll
        for (int t = 0; t < EGO_TILES; ++t) {
            v8f c = {};
            c = __builtin_amdgcn_wmma_f32_16x16x4_f32(
                    false, a[t], false, bb, (short)0, c, false, false);
            #pragma unroll
            for (int v = 0; v < 8; ++v) mn[t][v] = fmin_fast(mn[t][v], c[v]);
        }
    }

    // Row violates drivable iff NO lane point is within 3m:
    // OR-reduce "within 3m" bits, count zero bits.
    int driv_rows = 0;
    #pragma unroll
    for (int t = 0; t < EGO_TILES; ++t) {
        unsigned u = 0u;
        #pragma unroll
        for (int v = 0; v < 8; ++v)
            u |= (mn[t][v] <= LANE_R2) ? (1u << v) : 0u;
        #pragma unroll
        for (int s = 1; s <= 8; s <<= 1)
            u |= (unsigned)__shfl_xor((int)u, s, 32);
        int cnt = 8 - __popc(u);             // rows with min_dist > 3m
        cnt += __shfl_xor(cnt, 16, 32);
        driv_rows += cnt;
    }

    // ================= comfort (jerk) + progress =================
    float jsum = 0.0f;
    for (int i = lane; i < T_ - 3; i += 32) {      // 77 jerk samples
        const float x0 = tr[(i + 0) * 3], y0 = tr[(i + 0) * 3 + 1];
        const float x1 = tr[(i + 1) * 3], y1 = tr[(i + 1) * 3 + 1];
        const float x2 = tr[(i + 2) * 3], y2 = tr[(i + 2) * 3 + 1];
        const float x3 = tr[(i + 3) * 3], y3 = tr[(i + 3) * 3 + 1];
        const float jx = x3 - 3.0f * x2 + 3.0f * x1 - x0;
        const float jy = y3 - 3.0f * y2 + 3.0f * y1 - y0;
        jsum += sqrtf(jx * jx + jy * jy);
    }
    #pragma unroll
    for (int s = 1; s < 32; s <<= 1) jsum += __shfl_xor(jsum, s, 32);

    const float comfort   = -jsum / (float)(T_ - 3);
    const float progress  = tr[(T_ - 1) * 3 + 0];
    const float collision = -(float)coll_rows / (float)T_;
    const float drivable  = -(float)driv_rows / (float)T_;

    if (lane == 0)
        partial[idx] = W_COMFORT * comfort + W_PROGRESS * progress +
                       W_COLLISION * collision + W_DRIVABLE * drivable;
}

// One block per batch: softmax over modes, argmax, gather winning trajectory.
__global__ __launch_bounds__(64)
void select_kernel(const float* __restrict__ logits,   // B,M
                   const float* __restrict__ trajs,    // B,M,T,3
                   const float* __restrict__ partial,  // B*M
                   float* __restrict__ out)            // B*T*3 traj + B idx
{
    const int b   = blockIdx.x;
    const int tid = threadIdx.x;
    __shared__ float sc[M_];
    __shared__ int   bestIdx;

    if (tid < M_) sc[tid] = logits[b * M_ + tid];
    __syncthreads();

    if (tid == 0) {
        float mx = -3.0e38f;
        for (int m = 0; m < M_; ++m) mx = fmaxf(mx, sc[m]);
        float s = 0.0f;
        for (int m = 0; m < M_; ++m) s += expf(sc[m] - mx);
        const float inv = 1.0f / s;
        float bestS = -3.0e38f;
        int   bi    = 0;
        for (int m = 0; m < M_; ++m) {
            const float score = W_MODE * (expf(sc[m] - mx) * inv) +
                                partial[b * M_ + m];
            if (score > bestS) { bestS = score; bi = m; }   // first max wins
        }
        bestIdx = bi;
    }
    __syncthreads();

    const int bi = bestIdx;
    const float* src = trajs + (size_t)(b * M_ + bi) * (T_ * 3);
    for (int i = tid; i < T_ * 3; i += 64)
        out[b * T_ * 3 + i] = src[i];
    if (tid == 0)
        out[B_ * T_ * 3 + b] = (float)bi;
}

extern "C" void kernel_launch(void* const* d_in, const int* in_sizes, int n_in,
                              void* d_out, int out_size, void* d_ws, size_t ws_size,
                              hipStream_t stream) {
    const float* logits = (const float*)d_in[0];  // mode_logits  (B,M)
    const float* trajs  = (const float*)d_in[1];  // all_trajs    (B,M,T,3)
    const float* agents = (const float*)d_in[2];  // agents_now   (B,128,4)
    const float* amask  = (const float*)d_in[3];  // agents_mask  (B,128)
    const float* lanes  = (const float*)d_in[4];  // map_lanes    (B,64,20,3)
    const float* lmask  = (const float*)d_in[5];  // map_lanes_mask (B,64)

    float* partial = (float*)d_ws;                // B*M floats of scratch

    // 960 (b,m) pairs, 4 waves per 128-thread block -> 240 blocks
    score_modes_kernel<<<240, 128, 0, stream>>>(trajs, agents, amask,
                                                lanes, lmask, partial);
    select_kernel<<<B_, 64, 0, stream>>>(logits, trajs, partial, (float*)d_out);
}